// RetinaNetCriteriaV2_64261300682897
// MI455X (gfx1250) — compile-verified
//
#include <hip/hip_runtime.h>
#include <hip/hip_bf16.h>
#include <cstdint>
#include <math.h>

typedef __attribute__((ext_vector_type(2))) float v2f;
typedef __attribute__((ext_vector_type(8))) float v8f;

#define IGNORE_LABEL (-1.0f)
#define POS_THR 0.5f
#define NEG_THR 0.4f
#define BBOX_CLIP 4.135166556742356f   // log(1000/16)
#define NQ 7                            // cls_sum, npos, sl1_sum, fg_cnt, iou_sum, num_sum, valid_cnt
#define BLOCK 256

// ---- CDNA5 async global->LDS copy (gfx1250), tracked by ASYNCcnt ----
__device__ __forceinline__ void async_ld_to_lds_b32(uint32_t lds_byte_off, const float* gptr) {
  asm volatile("global_load_async_to_lds_b32 %0, %1, off"
               :: "v"(lds_byte_off), "v"(gptr)
               : "memory");
}
__device__ __forceinline__ void wait_asynccnt0() {
  asm volatile("s_wait_asynccnt 0" ::: "memory");
}

// WMMA-based sum of the 32 lane values 's' (wave32). Returns full sum in lane 0.
// A(16x4 f32, 2 VGPRs) holds the 32 values (+32 zeros), B = ones, so
// D[m][0] = rowsum(m); col 0 lives in lane 0 (M=0..7) and lane 16 (M=8..15).
__device__ __forceinline__ float wave_sum_wmma(float s) {
  v2f A; A.x = s; A.y = 0.0f;
  v2f B; B.x = 1.0f; B.y = 1.0f;
  v8f C = {};
  v8f D = __builtin_amdgcn_wmma_f32_16x16x4_f32(false, A, false, B, (short)0, C, false, false);
  float t = D[0] + D[1] + D[2] + D[3] + D[4] + D[5] + D[6] + D[7];
  return t + __shfl(t, 16, 32);
}

__global__ void __launch_bounds__(BLOCK)
retina_losses_kernel(const float* __restrict__ pred_cls,      // (n, A, 2) sigmoid probs
                     const float* __restrict__ rpn_num_prob,  // (n, A, 2) logits
                     const float* __restrict__ pred_reg,      // (n, A, 8)
                     const float* __restrict__ anchors,       // (A, 4)
                     const float* __restrict__ rpn_iou,       // (n, A, 2)
                     const float* __restrict__ boxes,         // (n, G, 5)
                     float* __restrict__ parts,               // (NQ, nblocks) per-block partials
                     int A, int G, int nblocks) {
  extern __shared__ float smem[];  // max(G*5, NQ*BLOCK) floats
  const int img = blockIdx.y;
  const int a   = blockIdx.x * BLOCK + threadIdx.x;
  const int tid = threadIdx.x;
  const int bid = blockIdx.y * gridDim.x + blockIdx.x;

  // ---- stage this image's GT boxes into LDS via async DMA path ----
  const float* gsrc = boxes + (size_t)img * G * 5;
  const int tot = G * 5;
  for (int i = tid; i < tot; i += BLOCK) {
    async_ld_to_lds_b32((uint32_t)(uintptr_t)&smem[i], gsrc + i);
  }
  wait_asynccnt0();
  __syncthreads();

  float cls_sum = 0.f, npos = 0.f, sl1_sum = 0.f, fg_cnt = 0.f;
  float iou_sum = 0.f, num_sum = 0.f, valid_cnt = 0.f;

  if (a < A) {
    const float ax0 = anchors[a * 4 + 0], ay0 = anchors[a * 4 + 1];
    const float ax1 = anchors[a * 4 + 2], ay1 = anchors[a * 4 + 3];
    const float aw = ax1 - ax0 + 1.f, ah = ay1 - ay0 + 1.f;
    const float area_a = aw * ah;
    const float acx = ax0 + 0.5f * aw, acy = ay0 + 0.5f * ah;

    // ---- pass A: top-2 IoU(anchor, gt), ignored gt forced to -1 ----
    float v1 = -2.f, v2 = -2.f; int i1 = 0, i2 = 0;
    for (int g = 0; g < G; ++g) {
      const float gx0 = smem[5*g], gy0 = smem[5*g+1], gx1 = smem[5*g+2], gy1 = smem[5*g+3];
      const float gl  = smem[5*g+4];
      const float iw = fminf(ax1, gx1) - fmaxf(ax0, gx0) + 1.f;
      const float ih = fminf(ay1, gy1) - fmaxf(ay0, gy0) + 1.f;
      const float inter = fmaxf(iw, 0.f) * fmaxf(ih, 0.f);
      const float area_g = (gx1 - gx0 + 1.f) * (gy1 - gy0 + 1.f);
      float ov = inter / fmaxf(area_a + area_g - inter, 1.f);
      if (gl == IGNORE_LABEL) ov = -1.f;
      if (ov > v1)      { v2 = v1; i2 = i1; v1 = ov; i1 = g; }
      else if (ov > v2) { v2 = ov; i2 = g; }
    }
    const float lab0 = (v1 >= POS_THR) ? 1.f : ((v1 < NEG_THR) ? 0.f : IGNORE_LABEL);
    const float lab1 = (v2 >= POS_THR) ? 1.f : ((v2 < NEG_THR) ? 0.f : IGNORE_LABEL);

    // ---- bbox_transform regression targets for both matched GTs ----
    float t0[4], t1[4];
    {
      const int gi[2] = { i1, i2 };
      float* tt[2] = { t0, t1 };
      for (int s = 0; s < 2; ++s) {
        const float gx0 = smem[5*gi[s]], gy0 = smem[5*gi[s]+1];
        const float gx1 = smem[5*gi[s]+2], gy1 = smem[5*gi[s]+3];
        const float gw = gx1 - gx0 + 1.f, gh = gy1 - gy0 + 1.f;
        const float gcx = gx0 + 0.5f * gw, gcy = gy0 + 0.5f * gh;
        tt[s][0] = (gcx - acx) / aw;
        tt[s][1] = (gcy - acy) / ah;
        tt[s][2] = logf(gw / aw);
        tt[s][3] = logf(gh / ah);
      }
    }

    // ---- pass B: decode both predicted boxes, IoU-target with exclusion ----
    float d[8];
    {
      const float* dr = pred_reg + ((size_t)img * A + a) * 8;
      #pragma unroll
      for (int k = 0; k < 8; ++k) d[k] = dr[k];
    }
    float bx[2][4], area_d[2];
    #pragma unroll
    for (int s = 0; s < 2; ++s) {
      const float pcx = acx + d[4*s + 0] * aw;
      const float pcy = acy + d[4*s + 1] * ah;
      const float pw  = aw * expf(fminf(d[4*s + 2], BBOX_CLIP));
      const float ph  = ah * expf(fminf(d[4*s + 3], BBOX_CLIP));
      bx[s][0] = pcx - 0.5f * pw; bx[s][1] = pcy - 0.5f * ph;
      bx[s][2] = pcx + 0.5f * pw; bx[s][3] = pcy + 0.5f * ph;
      area_d[s] = (bx[s][2] - bx[s][0] + 1.f) * (bx[s][3] - bx[s][1] + 1.f);
    }
    float amax = -1.f; int ia = 0;
    float b1 = -1.f, b2 = -1.f; int j1 = 0;
    for (int g = 0; g < G; ++g) {
      const float gx0 = smem[5*g], gy0 = smem[5*g+1], gx1 = smem[5*g+2], gy1 = smem[5*g+3];
      const float keep = (smem[5*g+4] == IGNORE_LABEL) ? 0.f : 1.f;
      const float area_g = (gx1 - gx0 + 1.f) * (gy1 - gy0 + 1.f);
      float ov[2];
      #pragma unroll
      for (int s = 0; s < 2; ++s) {
        const float iw = fminf(bx[s][2], gx1) - fmaxf(bx[s][0], gx0) + 1.f;
        const float ih = fminf(bx[s][3], gy1) - fmaxf(bx[s][1], gy0) + 1.f;
        const float inter = fmaxf(iw, 0.f) * fmaxf(ih, 0.f);
        ov[s] = (inter / fmaxf(area_d[s] + area_g - inter, 1.f)) * keep;
      }
      if (ov[0] > amax) { amax = ov[0]; ia = g; }
      if (ov[1] > b1)      { b2 = b1; b1 = ov[1]; j1 = g; }
      else if (ov[1] > b2) { b2 = ov[1]; }
    }
    const float aval = fmaxf(amax, 0.f);
    const float bval = fmaxf((j1 == ia) ? b2 : b1, 0.f);

    // ---- labels with "ignores" transform (kept as generic floats) ----
    const float a0 = lab0, b0 = lab1;
    const float ign = b0 - ((a0 == 0.f) ? 1.f : 0.f) * ((b0 != 0.f) ? 1.f : 0.f);
    const float labs[2] = { a0, ign };
    const float iou_t[2] = { aval, bval };

    const size_t base2 = ((size_t)img * A + a) * 2;
    #pragma unroll
    for (int s = 0; s < 2; ++s) {
      const float l = labs[s];
      const float p = pred_cls[base2 + s];
      // focal loss
      const float mask = (l != IGNORE_LABEL) ? 1.f : 0.f;
      const float vl = l * mask;
      const float onehot = (vl == 1.f) ? 1.f : 0.f;
      const float om = 1.f - p;
      const float pos = om * om * onehot * logf(p);
      const float neg = p * p * (1.f - onehot) * logf(om);
      cls_sum += -(0.25f * pos + 0.75f * neg) * mask;
      npos += (l > 0.f) ? 1.f : 0.f;
      // smooth L1 (sigma = 3)
      const float fg = (l != 0.f && l != IGNORE_LABEL) ? 1.f : 0.f;
      const float* tgt = (s == 0) ? t0 : t1;
      float sl = 0.f;
      #pragma unroll
      for (int k = 0; k < 4; ++k) {
        const float x = d[4*s + k] - tgt[k];
        const float axv = fabsf(x);
        sl += (axv < (1.f / 9.f)) ? 4.5f * x * x : axv - (1.f / 18.f);
      }
      sl1_sum += sl * fg;
      fg_cnt  += fg;
      // IoU L1
      iou_sum += fabsf(rpn_iou[base2 + s] - iou_t[s]) * fg;
    }
    // num loss: 2-class softmax CE on count-1 label
    const int nlab = ((a0 > 0.f) ? 1 : 0) + ((ign > 0.f) ? 1 : 0) - 1;
    if (nlab != -1) {
      const float s0 = rpn_num_prob[base2 + 0];
      const float s1 = rpn_num_prob[base2 + 1];
      const float m = fmaxf(s0, s1);
      const float lse = m + logf(expf(s0 - m) + expf(s1 - m));
      num_sum += -(((nlab == 0) ? s0 : s1) - lse);
      valid_cnt += 1.f;
    }
  }

  // ---- block reduction: LDS gather, then WMMA wave-sum in wave 0 ----
  __syncthreads();  // everyone done reading GT data from smem
  const float part[NQ] = { cls_sum, npos, sl1_sum, fg_cnt, iou_sum, num_sum, valid_cnt };
  #pragma unroll
  for (int q = 0; q < NQ; ++q) smem[q * BLOCK + tid] = part[q];
  __syncthreads();

  if (tid < 32) {
    for (int q = 0; q < NQ; ++q) {
      float s = 0.f;
      #pragma unroll
      for (int w = 0; w < BLOCK / 32; ++w) s += smem[q * BLOCK + tid + 32 * w];
      const float tot = wave_sum_wmma(s);   // full sum in lane 0
      if (tid == 0) parts[(size_t)q * nblocks + bid] = tot;
    }
  }
}

// Deterministic final reduction over per-block partials + loss assembly.
__global__ void __launch_bounds__(BLOCK)
finalize_kernel(const float* __restrict__ parts, int nblocks, float* __restrict__ out) {
  __shared__ float red[NQ][BLOCK];
  const int t = threadIdx.x;
  for (int q = 0; q < NQ; ++q) {
    float s = 0.f;
    for (int b = t; b < nblocks; b += BLOCK) s += parts[(size_t)q * nblocks + b];
    red[q][t] = s;
  }
  __syncthreads();
  for (int off = BLOCK / 2; off > 0; off >>= 1) {
    if (t < off) {
      #pragma unroll
      for (int q = 0; q < NQ; ++q) red[q][t] += red[q][t + off];
    }
    __syncthreads();
  }
  if (t == 0) {
    const float cls = red[0][0], np = red[1][0], sl1 = red[2][0];
    const float fg = red[3][0], iou = red[4][0], num = red[5][0], valid = red[6][0];
    out[0] = cls / fmaxf(np, 1.f);
    out[1] = 2.f * sl1 / fmaxf(fg, 1.f);
    out[2] = 2.f * iou / fmaxf(fg, 1.f);
    out[3] = num / fmaxf(valid, 1.f);
  }
}

extern "C" void kernel_launch(void* const* d_in, const int* in_sizes, int n_in,
                              void* d_out, int out_size, void* d_ws, size_t ws_size,
                              hipStream_t stream) {
  const float* pred_cls     = (const float*)d_in[0];  // (n, A, 2)
  const float* rpn_num_prob = (const float*)d_in[1];  // (n, A, 2)
  const float* pred_reg     = (const float*)d_in[2];  // (n, A, 8)
  const float* anchors      = (const float*)d_in[3];  // (A, 4)
  const float* rpn_iou      = (const float*)d_in[4];  // (n, A, 2)
  const float* boxes        = (const float*)d_in[5];  // (n, G, 5)
  // d_in[6] = im_info, unused

  const int A = in_sizes[3] / 4;
  const int n = in_sizes[0] / (A * 2);
  const int G = in_sizes[5] / (n * 5);

  const int gridX = (A + BLOCK - 1) / BLOCK;
  const int nblocks = gridX * n;
  dim3 grid(gridX, n);

  int smem_floats = G * 5;
  if (smem_floats < NQ * BLOCK) smem_floats = NQ * BLOCK;
  const size_t shmem = sizeof(float) * (size_t)smem_floats;

  float* parts = (float*)d_ws;  // NQ * nblocks floats (fully written each call)

  retina_losses_kernel<<<grid, BLOCK, shmem, stream>>>(
      pred_cls, rpn_num_prob, pred_reg, anchors, rpn_iou, boxes, parts, A, G, nblocks);
  finalize_kernel<<<1, BLOCK, 0, stream>>>(parts, nblocks, (float*)d_out);
}